// SegmentCSR_37151467111232
// MI455X (gfx1250) — compile-verified
//
#include <hip/hip_runtime.h>

// SegmentCSR sum via WMMA segment-assignment masking (gfx1250 / CDNA5).
//
// One wave32 owns 16 consecutive segments. Per iteration it ingests 8 rows of
// x (D=64) and issues 8x V_WMMA_F32_16X16X4_F32 where:
//   A (16x4)  = membership mask: A[m][k] = (row k in segment s0+m) ? 1 : 0
//   B (4x16)  = x rows for one 16-column group
//   C (16x16) = running segment sums (16 segments x 16 cols), 4 accumulators.
// Monotone indptr guarantees each row hits exactly one A-row; overshoot/tail
// rows mask to zero (A=0 x finite = 0), so the x2 unroll needs no tail loop.
// Loads are exactly-once (16 b32 per lane per 8-row chunk = 2KB/wave-iter),
// stores are coalesced 64B half-wave slices. Memory-bound: ~1.07GB traffic
// -> ~46us at 23.3 TB/s is the ceiling.

typedef __attribute__((ext_vector_type(2))) float v2f;
typedef __attribute__((ext_vector_type(8))) float v8f;

#define SEGS_PER_WAVE 16
#define FEAT 64
#define ROW_BYTES 256u   // FEAT * 4

__global__ __launch_bounds__(256) void segcsr_wmma_kernel(
    const float* __restrict__ x,
    const long long* __restrict__ indptr,
    float* __restrict__ out,
    int n_rows, int n_segs)
{
    const int lane = (int)(threadIdx.x & 31u);
    const int wave = (int)(blockIdx.x * (blockDim.x >> 5) + (threadIdx.x >> 5));
    const int s0   = wave * SEGS_PER_WAVE;
    if (s0 >= n_segs) return;   // wave-uniform exit

    const int m = lane & 15;    // A-matrix row (segment-within-tile) this lane feeds

    // Per-lane segment bounds for mask construction (clamped; rows of a
    // duplicated segment land in D-rows we never store).
    const int sm  = s0 + m;
    const int smc = (sm < n_segs) ? sm : (n_segs - 1);
    const int lo  = (int)indptr[smc];
    const int hi  = (int)indptr[smc + 1];

    // Wave-uniform row range covered by these 16 segments.
    const bool full = (s0 + SEGS_PER_WAVE) <= n_segs;
    const int  sEnd = full ? (s0 + SEGS_PER_WAVE) : n_segs;
    const int  base = (int)indptr[s0];
    const int  end  = (int)indptr[sEnd];

    v8f c0 = {}, c1 = {}, c2 = {}, c3 = {};

    // A 16x4 f32 layout: lanes 0-15 carry K=0 (v[0]) and K=1 (v[1]);
    // lanes 16-31 carry K=2 and K=3 (ISA 7.12.2).
    const int khalf = (lane >> 4) << 1;
    const char* xb = (const char*)x;
    const unsigned col_off = (unsigned)m * 4u;

    for (int cb = base; cb < end; cb += 8) {
        // ---- rows cb+0 .. cb+3 ----
        {
            const int k0 = cb + khalf;
            const int k1 = k0 + 1;
            v2f a;
            a.x = (k0 >= lo && k0 < hi) ? 1.0f : 0.0f;
            a.y = (k1 >= lo && k1 < hi) ? 1.0f : 0.0f;
            const int r0 = (k0 < n_rows) ? k0 : (n_rows - 1);
            const int r1 = (k1 < n_rows) ? k1 : (n_rows - 1);
            const char* p0 = xb + ((unsigned)r0 * ROW_BYTES + col_off);
            const char* p1 = xb + ((unsigned)r1 * ROW_BYTES + col_off);
            v2f b0 = { *(const float*)(p0 +   0), *(const float*)(p1 +   0) };
            v2f b1 = { *(const float*)(p0 +  64), *(const float*)(p1 +  64) };
            v2f b2 = { *(const float*)(p0 + 128), *(const float*)(p1 + 128) };
            v2f b3 = { *(const float*)(p0 + 192), *(const float*)(p1 + 192) };
            c0 = __builtin_amdgcn_wmma_f32_16x16x4_f32(false, a, false, b0, (short)0, c0, false, false);
            c1 = __builtin_amdgcn_wmma_f32_16x16x4_f32(false, a, false, b1, (short)0, c1, false, false);
            c2 = __builtin_amdgcn_wmma_f32_16x16x4_f32(false, a, false, b2, (short)0, c2, false, false);
            c3 = __builtin_amdgcn_wmma_f32_16x16x4_f32(false, a, false, b3, (short)0, c3, false, false);
        }
        // ---- rows cb+4 .. cb+7 (mask kills any overshoot past end) ----
        {
            const int k0 = cb + 4 + khalf;
            const int k1 = k0 + 1;
            v2f a;
            a.x = (k0 >= lo && k0 < hi) ? 1.0f : 0.0f;
            a.y = (k1 >= lo && k1 < hi) ? 1.0f : 0.0f;
            const int r0 = (k0 < n_rows) ? k0 : (n_rows - 1);
            const int r1 = (k1 < n_rows) ? k1 : (n_rows - 1);
            const char* p0 = xb + ((unsigned)r0 * ROW_BYTES + col_off);
            const char* p1 = xb + ((unsigned)r1 * ROW_BYTES + col_off);
            v2f b0 = { *(const float*)(p0 +   0), *(const float*)(p1 +   0) };
            v2f b1 = { *(const float*)(p0 +  64), *(const float*)(p1 +  64) };
            v2f b2 = { *(const float*)(p0 + 128), *(const float*)(p1 + 128) };
            v2f b3 = { *(const float*)(p0 + 192), *(const float*)(p1 + 192) };
            c0 = __builtin_amdgcn_wmma_f32_16x16x4_f32(false, a, false, b0, (short)0, c0, false, false);
            c1 = __builtin_amdgcn_wmma_f32_16x16x4_f32(false, a, false, b1, (short)0, c1, false, false);
            c2 = __builtin_amdgcn_wmma_f32_16x16x4_f32(false, a, false, b2, (short)0, c2, false, false);
            c3 = __builtin_amdgcn_wmma_f32_16x16x4_f32(false, a, false, b3, (short)0, c3, false, false);
        }
    }

    // C/D 16x16 f32 layout: VGPR i -> (M=i, N=lane) for lanes 0-15,
    // (M=i+8, N=lane-16) for lanes 16-31. Store rows = segments, cols = feats.
    const int mrow = (lane >> 4) << 3;  // +8 for upper half-wave
    const int ncol = lane & 15;
    char* ob = (char*)out + ((unsigned)(s0 + mrow) * ROW_BYTES + (unsigned)ncol * 4u);

    if (full) {
        // Wave-uniform fast path: straight-line coalesced stores.
        #pragma unroll
        for (int i = 0; i < 8; ++i) {
            char* o = ob + (unsigned)i * ROW_BYTES;
            *(float*)(o +   0) = c0[i];
            *(float*)(o +  64) = c1[i];
            *(float*)(o + 128) = c2[i];
            *(float*)(o + 192) = c3[i];
        }
    } else {
        #pragma unroll
        for (int i = 0; i < 8; ++i) {
            if (s0 + mrow + i < n_segs) {
                char* o = ob + (unsigned)i * ROW_BYTES;
                *(float*)(o +   0) = c0[i];
                *(float*)(o +  64) = c1[i];
                *(float*)(o + 128) = c2[i];
                *(float*)(o + 192) = c3[i];
            }
        }
    }
}

extern "C" void kernel_launch(void* const* d_in, const int* in_sizes, int n_in,
                              void* d_out, int out_size, void* d_ws, size_t ws_size,
                              hipStream_t stream) {
    const float*     x      = (const float*)d_in[0];
    const long long* indptr = (const long long*)d_in[1];  // int64 per reference
    float*           out    = (float*)d_out;

    const int n_rows = in_sizes[0] / FEAT;      // 4194304
    const int n_segs = in_sizes[1] - 1;         // 262144

    const int waves   = (n_segs + SEGS_PER_WAVE - 1) / SEGS_PER_WAVE; // 16384
    const int threads = 256;                                          // 8 waves
    const int blocks  = (waves * 32 + threads - 1) / threads;         // 2048

    segcsr_wmma_kernel<<<blocks, threads, 0, stream>>>(x, indptr, out, n_rows, n_segs);
}